// RegionLoss_19318762897784
// MI455X (gfx1250) — compile-verified
//
#include <hip/hip_runtime.h>
#include <math.h>

// Problem constants (from reference)
#define NB   32
#define NANC 5
#define NCH_A 32            // 19 + NC(13) channels per anchor
#define NH   76
#define NW   76
#define NBOX 50
#define PLANE (NH*NW)       // 5776
#define NCELLS (NB*PLANE)   // 184832
#define NCHTOT (NANC*NCH_A) // 160
#define MAIN_BLOCK 256
#define MAIN_GRID  (NCELLS/MAIN_BLOCK)  // 722 exactly, no partial block

typedef float v2f __attribute__((ext_vector_type(2)));
typedef float v8f __attribute__((ext_vector_type(8)));

// sigmoid with hardware reciprocal (v_rcp_f32, ~1 ulp) instead of IEEE divide:
// saves the ~10-op v_div_fmas/v_div_fixup ladder on the hot path.
__device__ __forceinline__ float sigmoidf(float x) {
    float e = expf(-x);
#if __has_builtin(__builtin_amdgcn_rcpf)
    return __builtin_amdgcn_rcpf(1.0f + e);
#else
    return 1.0f / (1.0f + e);
#endif
}

// Exact f32 wave(32) reduction using the matrix pipe:
// D = ones(16x4) * B(4x16) + 0  ->  every D element = column-sum of B.
// Each lane contributes one value (b.x); the 64 B slots (32 lanes x 2 VGPRs)
// hold each lane's value exactly once (b.y = 0). d[0] for lane L is
// colsum(L % 16): each 16-lane half holds all 16 distinct column sums, so
// 4 xor-shuffles confined to a half produce the exact grand total.
__device__ __forceinline__ float wave_reduce_sum(float v) {
#if __has_builtin(__builtin_amdgcn_wmma_f32_16x16x4_f32)
    v2f a; a.x = 1.0f; a.y = 1.0f;        // A = all-ones 16x4
    v2f b; b.x = v;    b.y = 0.0f;        // B carries one value per lane
    v8f c = {0.f,0.f,0.f,0.f,0.f,0.f,0.f,0.f};
    v8f d = __builtin_amdgcn_wmma_f32_16x16x4_f32(
        /*neg_a=*/false, a, /*neg_b=*/false, b,
        /*c_mod=*/(short)0, c, /*reuse_a=*/false, /*reuse_b=*/false);
    float col = d[0];
    col += __shfl_xor(col, 8, 32);
    col += __shfl_xor(col, 4, 32);
    col += __shfl_xor(col, 2, 32);
    col += __shfl_xor(col, 1, 32);
    return col;
#else
    for (int off = 16; off >= 1; off >>= 1) v += __shfl_xor(v, off, 32);
    return v;
#endif
}

// ---------------- kernel 1a: init scatter map to -1 ----------------
__global__ void rl_init_map(int* __restrict__ map) {
    int idx = blockIdx.x * blockDim.x + threadIdx.x;
    if (idx < NCELLS) map[idx] = -1;
}

// ---------------- kernel 1b: sequential last-wins scatter ----------------
// One lane per batch; walks boxes 0..49 in order (matches sequential
// "last write wins" semantics and the cumprod validity chain).
__global__ void rl_scatter(const float* __restrict__ tgt, int* __restrict__ map) {
    int b = (int)threadIdx.x;
    if (b >= NB) return;
    const float* tb = tgt + (size_t)b * NBOX * 21;
    int valid = 1;
    for (int t = 0; t < NBOX; ++t) {
        float x = tb[t * 21 + 1];
        float y = tb[t * 21 + 2];
        if (x == 0.0f) valid = 0;
        if (!valid) continue;
        int gi = (int)floorf(x * (float)NW);
        int gj = (int)floorf(y * (float)NH);
        if (gi >= 0 && gi < NW && gj >= 0 && gj < NH)
            map[b * PLANE + gj * NW + gi] = t;   // last valid box wins
    }
}

// ---------------- kernel 2: main loss ----------------
__global__ __launch_bounds__(MAIN_BLOCK)
void rl_main(const float* __restrict__ out, const float* __restrict__ tgt,
             const int* __restrict__ map, float* __restrict__ partials) {
    const int cell = blockIdx.x * MAIN_BLOCK + threadIdx.x;   // always < NCELLS
    const int b = cell / PLANE;
    const int r = cell - b * PLANE;
    const int j = r / NW;
    const int i = r - j * NW;

    // base index of channel 0 for this (b, j, i)
    const size_t baseCell = (size_t)b * NCHTOT * PLANE + (size_t)j * NW + i;

    float loss = 0.0f;
    const int t = map[b * PLANE + r];

    float mask0 = 1.0f, tconf0 = 0.0f;

    if (t >= 0) {
        // gather 18 coord channels of anchor 0 at this cell (sigmoid on 0,1)
        float p[18];
        #pragma unroll
        for (int c = 0; c < 18; ++c) {
            float v = out[baseCell + (size_t)c * PLANE];
            p[c] = (c < 2) ? sigmoidf(v) : v;
        }
        const float* tc = tgt + (size_t)b * NBOX * 21 + (size_t)t * 21 + 1;
        float coords[18];
        #pragma unroll
        for (int c = 0; c < 18; ++c) coords[c] = tc[c];

        // conf_t from point distances (stop-gradient path)
        float s = 0.0f;
        #pragma unroll
        for (int k = 0; k < 9; ++k) {
            float px = (p[2*k]     + (float)i) * (1.0f / (float)NW);
            float py = (p[2*k + 1] + (float)j) * (1.0f / (float)NH);
            float dX = (coords[2*k]     - px) * 640.0f;
            float dY = (coords[2*k + 1] - py) * 480.0f;
            float dn = sqrtf(dX * dX + dY * dY);
            if (dn < 80.0f)
                s += expf(2.0f * (1.0f - dn * (1.0f / 80.0f))) - 1.0f;
        }
        tconf0 = (s * (1.0f / 9.0f)) / (expf(2.0f) - 1.0f + 1e-5f);
        mask0  = 5.0f;   // OBJ_SCALE

        // coord (xy) loss: masked cell, anchor 0, channels 0..17
        #pragma unroll
        for (int c = 0; c < 18; ++c) {
            float tar = coords[c] * 76.0f - (float)((c & 1) ? j : i);
            float dlt = p[c] - tar;
            loss += 0.5f * dlt * dlt;   // COORD_SCALE = 1
        }
    }

    // conf loss, all 5 anchors (obj mask only possible at anchor 0)
    {
        float x0 = out[baseCell + (size_t)18 * PLANE];
        float d0 = sigmoidf(x0) - tconf0;
        loss += 0.5f * mask0 * d0 * d0;
    }
    #pragma unroll
    for (int a = 1; a < NANC; ++a) {
        float xa = out[baseCell + (size_t)(a * NCH_A + 18) * PLANE];
        float ca = sigmoidf(xa);
        loss += 0.5f * ca * ca;        // NOOBJ_SCALE = 1, tconf = 0
    }

    // block reduction: WMMA wave-sum, then LDS across the 8 waves
    float wsum = wave_reduce_sum(loss);
    __shared__ float smem[MAIN_BLOCK / 32];
    if ((threadIdx.x & 31u) == 0) smem[threadIdx.x >> 5] = wsum;
    __syncthreads();
    if (threadIdx.x == 0) {
        float tot = 0.0f;
        #pragma unroll
        for (int w = 0; w < MAIN_BLOCK / 32; ++w) tot += smem[w];
        partials[blockIdx.x] = tot;
    }
}

// ---------------- kernel 3: deterministic finalize (1 wave) ----------------
__global__ void rl_finalize(const float* __restrict__ partials, int n,
                            float* __restrict__ outp) {
    float acc = 0.0f;
    for (int idx = (int)threadIdx.x; idx < n; idx += 32) acc += partials[idx];
    float tot = wave_reduce_sum(acc);
    if (threadIdx.x == 0) outp[0] = tot;
}

extern "C" void kernel_launch(void* const* d_in, const int* in_sizes, int n_in,
                              void* d_out, int out_size, void* d_ws, size_t ws_size,
                              hipStream_t stream) {
    (void)in_sizes; (void)n_in; (void)out_size; (void)ws_size;
    const float* out = (const float*)d_in[0];   // (32, 160, 76, 76) f32
    const float* tgt = (const float*)d_in[1];   // (32, 1050) f32

    // workspace layout: int map[NCELLS] | float partials[MAIN_GRID]
    int*   map      = (int*)d_ws;
    float* partials = (float*)((char*)d_ws + (size_t)NCELLS * sizeof(int));

    rl_init_map<<<(NCELLS + 255) / 256, 256, 0, stream>>>(map);
    rl_scatter<<<1, 32, 0, stream>>>(tgt, map);
    rl_main<<<MAIN_GRID, MAIN_BLOCK, 0, stream>>>(out, tgt, map, partials);
    rl_finalize<<<1, 32, 0, stream>>>(partials, MAIN_GRID, (float*)d_out);
}